// MMCL_77859167141895
// MI455X (gfx1250) — compile-verified
//
#include <hip/hip_runtime.h>
#include <math.h>
#include <stdint.h>

#define B_ROWS   256
#define N_COLS   65536
#define DELTA_C  5.0f
#define BLK      1024
#define NBINS    4096
#define CAP      2048
#define MAXPOS   64

typedef unsigned int u32x4 __attribute__((ext_vector_type(4)));
typedef int          i32x8 __attribute__((ext_vector_type(8)));
typedef int          i32x4 __attribute__((ext_vector_type(4)));

__device__ __forceinline__ float softplusf(float x) {
    // stable: max(x,0) + log1p(exp(-|x|))
    return fmaxf(x, 0.0f) + log1pf(expf(-fabsf(x)));
}

__device__ __forceinline__ unsigned fkey(float f) {
    // order-preserving map: larger float -> larger unsigned
    unsigned u = __float_as_uint(f);
    return (u & 0x80000000u) ? ~u : (u | 0x80000000u);
}

__global__ __launch_bounds__(BLK) void mmcl_row_kernel(
    const float* __restrict__ logits,
    const unsigned char* __restrict__ targets,
    const float* __restrict__ co_cnts,
    float* __restrict__ row_out)
{
    // dynamic LDS: whole 256KB logits row, DMA'd in by the TDM
    extern __shared__ float4 row_lds4[];
    float* row_lds = (float*)row_lds4;

    __shared__ unsigned hist[NBINS];
    __shared__ float    cand[CAP];
    __shared__ float    red[BLK];
    __shared__ unsigned super[64];
    __shared__ int      poslist[MAXPOS];
    __shared__ int      s_npos, s_candcnt, s_m, s_refine, s_bbin, s_chi;
    __shared__ unsigned s_thresh;
    __shared__ float    s_posum;

    const int row = blockIdx.x;
    const int tid = threadIdx.x;
    const float*         lrow = logits  + (size_t)row * N_COLS;
    const unsigned char* trow = targets + (size_t)row * N_COLS;
    const float*         crow = co_cnts + (size_t)row * N_COLS;

    for (int i = tid; i < NBINS; i += BLK) hist[i] = 0u;
    if (tid == 0) { s_npos = 0; s_candcnt = 0; }

    // ---------- issue TDM: DMA whole logits row (4096x16 f32 tile) into LDS ----------
    if (tid == 0) {
        unsigned lds_off = (unsigned)(uintptr_t)row_lds;   // LDS byte offset = low 32 bits
        unsigned long long ga = (unsigned long long)(uintptr_t)lrow;
        u32x4 g0 = { 1u,                                   // count=1 (valid user descriptor)
                     lds_off,                               // lds_addr
                     (unsigned)ga,                          // global_addr[31:0]
                     (unsigned)((ga >> 32) & 0x01FFFFFFull) // global_addr[56:32]
                         | (2u << 30) };                    // type=2 ("image")
        i32x8 g1 = { (int)0x00020000,   // data_size=2 (4B); mask/flags/pad = 0
                     (int)0x10000000,   // tensor_dim0 = 4096  (bits[63:48])
                     (int)0x00100000,   // tensor_dim1 = 16    (bits[95:80])
                     (int)0x10000000,   // tile_dim0   = 4096  (bits[127:112])
                     (int)16,           // tile_dim1   = 16, tile_dim2 = 0
                     (int)4096,         // tensor_dim0_stride = 4096 (lo32)
                     0, 0 };            // stride hi / tensor_dim1_stride = 0
        i32x4 g2 = { 0, 0, 0, 0 };
        i32x4 g3 = { 0, 0, 0, 0 };
        i32x8 gx = { 0, 0, 0, 0, 0, 0, 0, 0 };  // extra arg of the 6-arg clang-23 form
        __builtin_amdgcn_tensor_load_to_lds(g0, g1, g2, g3, gx, 0);
    }
    __syncthreads();   // s_npos/hist init visible before atomics below

    // ---------- overlap: stream targets from global while TDM fills LDS ----------
    const unsigned* t4 = (const unsigned*)trow;
    for (int i = tid; i < N_COLS / 4; i += BLK) {
        unsigned tm = t4[i];
        if (tm) {  // rare: ~8 positives per 65536
            #pragma unroll
            for (int c = 0; c < 4; ++c) {
                if ((tm >> (8 * c)) & 0xFFu) {
                    int slot = atomicAdd(&s_npos, 1);
                    if (slot < MAXPOS) poslist[slot] = i * 4 + c;
                }
            }
        }
    }
    if (tid == 0) __builtin_amdgcn_s_wait_tensorcnt(0);    // TDM done -> LDS valid
    __syncthreads();

    // ---------- histogram of 12-bit key prefix, read from LDS ----------
    for (int i = tid; i < N_COLS / 4; i += BLK) {
        float4 v = row_lds4[i];
        atomicAdd(&hist[fkey(v.x) >> 20], 1u);
        atomicAdd(&hist[fkey(v.y) >> 20], 1u);
        atomicAdd(&hist[fkey(v.z) >> 20], 1u);
        atomicAdd(&hist[fkey(v.w) >> 20], 1u);
    }
    __syncthreads();
    if (tid < 64) {
        unsigned ss = 0;
        for (int j = 0; j < 64; ++j) ss += hist[tid * 64 + j];
        super[tid] = ss;
    }
    __syncthreads();

    // ---------- thread 0: positives (sorted, deterministic), k, m, boundary ----------
    if (tid == 0) {
        int npos = s_npos; if (npos > MAXPOS) npos = MAXPOS;
        // sort poslist ascending so l_pos sum order is replay-deterministic
        for (int a = 1; a < npos; ++a) {
            int v = poslist[a]; int b = a - 1;
            while (b >= 0 && poslist[b] > v) { poslist[b + 1] = poslist[b]; --b; }
            poslist[b + 1] = v;
        }
        float posum = 0.0f;
        for (int p = 0; p < npos; ++p) {
            int idx = poslist[p];
            posum += crow[idx] * softplusf(-row_lds[idx]);
        }
        s_posum = posum;

        int k = (int)floorf(0.01f * (float)(N_COLS - s_npos)); // f32, matches jnp weak typing
        int m = k;
        for (int it = 0; it < MAXPOS + 2; ++it) {
            int cnt = 0;
            for (int p = 0; p < npos; ++p) if (poslist[p] < m) cnt++;
            int nm = k + cnt;
            if (nm == m) break;
            m = nm;
        }
        if (m > CAP) m = CAP;  // safety clamp (cannot trigger for this problem)
        s_m = m;

        if (m > 0) {
            unsigned cum = 0; int sb = 63;
            for (; sb > 0; --sb) {
                if (cum + super[sb] >= (unsigned)m) break;
                cum += super[sb];
            }
            int b = sb * 64 + 63;
            for (; b > sb * 64; --b) {
                if (cum + hist[b] >= (unsigned)m) break;
                cum += hist[b];
            }
            s_bbin   = b;
            s_chi    = (int)cum;                       // strictly-above count
            s_refine = ((cum + hist[b]) > CAP) ? 1 : 0;
            s_thresh = ((unsigned)b) << 12;            // compare against key>>8
        } else {
            s_bbin = NBINS - 1; s_chi = 0; s_refine = 0;
            s_thresh = 0xFFFFFFFFu;                    // select nothing
        }
    }
    __syncthreads();

    // ---------- rare refine (pathologically dense boundary bin), from LDS ----------
    if (s_refine) {
        for (int i = tid; i < NBINS; i += BLK) hist[i] = 0u;
        __syncthreads();
        const int b = s_bbin;
        for (int i = tid; i < N_COLS; i += BLK) {
            unsigned key = fkey(row_lds[i]);
            if ((int)(key >> 20) == b) atomicAdd(&hist[(key >> 8) & 0xFFFu], 1u);
        }
        __syncthreads();
        if (tid < 64) {
            unsigned ss = 0;
            for (int j = 0; j < 64; ++j) ss += hist[tid * 64 + j];
            super[tid] = ss;
        }
        __syncthreads();
        if (tid == 0) {
            int m2 = s_m - s_chi;
            unsigned cum = 0; int sb = 63;
            for (; sb > 0; --sb) {
                if (cum + super[sb] >= (unsigned)m2) break;
                cum += super[sb];
            }
            int b2 = sb * 64 + 63;
            for (; b2 > sb * 64; --b2) {
                if (cum + hist[b2] >= (unsigned)m2) break;
                cum += hist[b2];
            }
            s_thresh = (((unsigned)s_bbin) << 12) | (unsigned)b2;
        }
        __syncthreads();
    }

    // ---------- collect candidates >= threshold, from LDS ----------
    const unsigned thr = s_thresh;
    for (int i = tid; i < N_COLS / 4; i += BLK) {
        float4 v = row_lds4[i];
        float lv[4] = {v.x, v.y, v.z, v.w};
        #pragma unroll
        for (int c = 0; c < 4; ++c) {
            if ((fkey(lv[c]) >> 8) >= thr) {
                int slot = atomicAdd(&s_candcnt, 1);
                if (slot < CAP) cand[slot] = lv[c];
            }
        }
    }
    __syncthreads();
    {
        int nc = s_candcnt; if (nc > CAP) nc = CAP;
        for (int i = tid; i < CAP; i += BLK)
            if (i >= nc) cand[i] = -INFINITY;
    }
    __syncthreads();

    // ---------- bitonic sort, descending, 2048 elems in LDS ----------
    for (unsigned size = 2; size <= CAP; size <<= 1) {
        for (unsigned strd = size >> 1; strd > 0; strd >>= 1) {
            for (unsigned i = tid; i < CAP; i += BLK) {
                unsigned j = i ^ strd;
                if (j > i) {
                    float a = cand[i], b = cand[j];
                    bool descBlock = ((i & size) == 0);
                    if ((a < b) == descBlock) { cand[i] = b; cand[j] = a; }
                }
            }
            __syncthreads();
        }
    }

    // ---------- l_neg: ranks j < m whose POSITION j is a negative ----------
    float lneg_part = 0.0f;
    const int m = s_m;
    for (int j = tid; j < m; j += BLK) {
        if (trow[j] == 0) lneg_part += softplusf(cand[j]);
    }
    red[tid] = lneg_part;
    __syncthreads();
    for (int s = BLK / 2; s > 0; s >>= 1) {
        if (tid < s) red[tid] += red[tid + s];
        __syncthreads();
    }

    if (tid == 0) {
        int npos = s_npos;
        int k = (int)floorf(0.01f * (float)(N_COLS - npos));
        float n_sel = (float)((k < 1) ? 1 : k);
        float l_neg = red[0] / n_sel;
        float n_pos = (float)((npos < 1) ? 1 : npos);
        float l_pos = s_posum / n_pos;
        row_out[row] = DELTA_C * l_pos + l_neg;
    }
}

__global__ __launch_bounds__(B_ROWS) void mmcl_reduce_kernel(
    const float* __restrict__ row_vals, float* __restrict__ out)
{
    __shared__ float red[B_ROWS];
    const int tid = threadIdx.x;
    red[tid] = row_vals[tid];
    __syncthreads();
    for (int s = B_ROWS / 2; s > 0; s >>= 1) {
        if (tid < s) red[tid] += red[tid + s];
        __syncthreads();
    }
    if (tid == 0) out[0] = red[0] * (1.0f / (float)B_ROWS);
}

extern "C" void kernel_launch(void* const* d_in, const int* in_sizes, int n_in,
                              void* d_out, int out_size, void* d_ws, size_t ws_size,
                              hipStream_t stream) {
    const float*         logits  = (const float*)d_in[0];
    const unsigned char* targets = (const unsigned char*)d_in[1];  // jnp bool = 1 byte
    const float*         co      = (const float*)d_in[2];
    float* rows = (float*)d_ws;   // 256 floats of scratch

    size_t dyn_lds = (size_t)N_COLS * sizeof(float);   // 256KB row tile (<=320KB WGP LDS)
    mmcl_row_kernel<<<B_ROWS, BLK, dyn_lds, stream>>>(logits, targets, co, rows);
    mmcl_reduce_kernel<<<1, B_ROWS, 0, stream>>>(rows, (float*)d_out);
}